// DiffusionPropagate_9216999817894
// MI455X (gfx1250) — compile-verified
//
#include <hip/hip_runtime.h>
#include <stdint.h>
#include <stddef.h>

#define N      4096
#define B      8
#define NITER  4

#define CHUNKS   32                 // s-dimension split
#define CHUNK_S  (N / CHUNKS)       // 128 source values per chunk
#define THREADS  256                // 1 column per thread
#define COL_TILES (N / THREADS)     // 16

typedef float v2f __attribute__((ext_vector_type(2)));
typedef float v4f __attribute__((ext_vector_type(4)));

typedef __attribute__((address_space(1))) int* gb32_ptr;
typedef __attribute__((address_space(3))) int* lb32_ptr;

// ---------------------------------------------------------------------------
// CDNA5 async global->LDS (ASYNCcnt). b32 variant: per-lane LDS destination
// addresses let the async unit do a transposing scatter for free.
// ---------------------------------------------------------------------------
__device__ __forceinline__ void async_copy_b32(const float* gsrc, float* lds_dst) {
#if __has_builtin(__builtin_amdgcn_global_load_async_to_lds_b32)
    __builtin_amdgcn_global_load_async_to_lds_b32(
        (gb32_ptr)gsrc, (lb32_ptr)lds_dst, /*imm offset*/0, /*imm cpol*/0);
#else
    *lds_dst = *gsrc;                       // synchronous fallback
#endif
}

__device__ __forceinline__ void wait_async_zero() {
#if __has_builtin(__builtin_amdgcn_global_load_async_to_lds_b32)
  #if __has_builtin(__builtin_amdgcn_s_wait_asynccnt)
    __builtin_amdgcn_s_wait_asynccnt(0);
  #else
    asm volatile("s_wait_asynccnt 0" ::: "memory");
  #endif
#endif
}

// ---------------------------------------------------------------------------
// Stage 1: partial[chunk][b][d] = prod_{s in chunk} (1 - pred[b][s]*P[s][d])
// 1 column/thread; accumulators packed over batch pairs (v2f -> v_pk_*_f32).
// pred chunk staged transposed in LDS: predT[s_local][b].
// ---------------------------------------------------------------------------
__global__ __launch_bounds__(THREADS)
void diff_stage1(const float* __restrict__ pred,
                 const float* __restrict__ P,
                 float* __restrict__ partial)
{
    __shared__ float predT[CHUNK_S * B];    // 4 KB, layout [j][b]

    const int t     = threadIdx.x;
    const int tile  = blockIdx.x;           // column tile
    const int chunk = blockIdx.y;           // s chunk
    const int s0    = chunk * CHUNK_S;

    // Transposing async scatter: 1024 b32 elements, 4 per lane.
    // e = b*128 + j ; global pred[b][s0+j] -> LDS predT[j][b]
#pragma unroll
    for (int k = 0; k < 4; ++k) {
        const int e = t + k * THREADS;      // 0..1023
        const int b = e >> 7;               // 0..7
        const int j = e & (CHUNK_S - 1);    // 0..127
        async_copy_b32(pred + (size_t)b * N + s0 + j, &predT[j * B + b]);
    }
    wait_async_zero();
    __syncthreads();

    const int d = tile * THREADS + t;       // this thread's column

    v2f acc[4];                             // batch pairs {0,1}{2,3}{4,5}{6,7}
#pragma unroll
    for (int i = 0; i < 4; ++i) acc[i] = (v2f){1.0f, 1.0f};

    const float* __restrict__ Pd = P + (size_t)s0 * N + d;

#pragma unroll 4
    for (int j = 0; j < CHUNK_S; ++j) {
        const float pv = Pd[(size_t)j * N];         // coalesced b32
        const v2f pvv = {pv, pv};                   // splat, amortized over 4 pairs

        // predT row j: 8 floats = 2 x ds_load_b128 (wave-broadcast)
        const v4f lo = *(const v4f*)&predT[j * B + 0];
        const v4f hi = *(const v4f*)&predT[j * B + 4];
        v2f pb[4];
        pb[0] = __builtin_shufflevector(lo, lo, 0, 1);
        pb[1] = __builtin_shufflevector(lo, lo, 2, 3);
        pb[2] = __builtin_shufflevector(hi, hi, 0, 1);
        pb[3] = __builtin_shufflevector(hi, hi, 2, 3);

#pragma unroll
        for (int i = 0; i < 4; ++i) {
            // acc *= (1 - pb*pv)  ==  fma(-(acc*pb), pv, acc)
            // fneg folds into VOP3P neg modifiers; v2f ops -> v_pk_mul/fma_f32
            acc[i] = __builtin_elementwise_fma(-(acc[i] * pb[i]), pvv, acc[i]);
        }
    }

    // partial[chunk][b][d]
    float* out = partial + (size_t)chunk * (B * N) + d;
#pragma unroll
    for (int i = 0; i < 4; ++i) {
        out[(size_t)(2 * i) * N]     = acc[i].x;
        out[(size_t)(2 * i + 1) * N] = acc[i].y;
    }
}

// ---------------------------------------------------------------------------
// Stage 2: pred_next[b][d] = 1 - prod_c partial[c][b][d]
// ---------------------------------------------------------------------------
__global__ __launch_bounds__(256)
void diff_stage2(const float* __restrict__ partial,
                 float* __restrict__ out)
{
    const int idx = blockIdx.x * 256 + threadIdx.x;   // flat (b,d), 0..B*N-1
    float prod = 1.0f;
#pragma unroll
    for (int c = 0; c < CHUNKS; ++c)
        prod *= partial[(size_t)c * (B * N) + idx];
    out[idx] = 1.0f - prod;
}

// ---------------------------------------------------------------------------
// Launch: 4 iterations, each = stage1 + stage2, ping-pong pred in d_ws.
// ws layout: [0, 4MB) partials | then predA (128KB) | predB (128KB)
// ---------------------------------------------------------------------------
extern "C" void kernel_launch(void* const* d_in, const int* in_sizes, int n_in,
                              void* d_out, int out_size, void* d_ws, size_t ws_size,
                              hipStream_t stream)
{
    (void)in_sizes; (void)n_in; (void)out_size; (void)ws_size;

    const float* preds = (const float*)d_in[0];   // [B, N] fp32
    // d_in[1] = seed_idx (int64) : unused by the reference math
    const float* P     = (const float*)d_in[2];   // [N, N] fp32

    char*  ws      = (char*)d_ws;
    float* partial = (float*)ws;                                   // CHUNKS*B*N f32 = 4 MB
    float* predA   = (float*)(ws + (size_t)CHUNKS * B * N * sizeof(float));
    float* predB   = predA + (size_t)B * N;

    const dim3 g1(COL_TILES, CHUNKS);   // 16 x 32 = 512 blocks (4096 waves)
    const dim3 b1(THREADS);
    const int  g2 = (B * N) / 256;      // 128 blocks

    const float* cur = preds;
    for (int it = 0; it < NITER; ++it) {
        diff_stage1<<<g1, b1, 0, stream>>>(cur, P, partial);
        float* nxt = (it == NITER - 1) ? (float*)d_out
                                       : ((it & 1) ? predB : predA);
        diff_stage2<<<g2, 256, 0, stream>>>(partial, nxt);
        cur = nxt;
    }
}